// GNN_18021682774977
// MI455X (gfx1250) — compile-verified
//
#include <hip/hip_runtime.h>
#include <math.h>

typedef __attribute__((ext_vector_type(2))) float v2f;
typedef __attribute__((ext_vector_type(8))) float v8f;

#define B_   8
#define C_   256
#define C2_  512
#define N_   2304   // 48*48
#define TOPK 16

// ---------------------------------------------------------------------------
// Kernel 1: featpos = tokens @ W^T + bias      featpos : (B, N, 2C)
//   x  : (B, C, N)   tokens[b,n,c] = x[b,c,n]
//   Wm : (2C, C)
// One wave computes a 16x32 tile (two 16x16 WMMA tiles sharing the A
// fragment); block = 4 waves -> 16 rows x 128 cols.  No divergent epilogue.
// ---------------------------------------------------------------------------
__global__ __launch_bounds__(128) void featpos_gemm(
    const float* __restrict__ x,
    const float* __restrict__ Wm,
    const float* __restrict__ bias,
    float* __restrict__ featpos)
{
    const int tid  = threadIdx.x;
    const int lane = tid & 31;
    const int wv   = tid >> 5;
    const int bz   = blockIdx.z;
    const int n0   = blockIdx.y * 16;                 // token rows
    const int o0   = blockIdx.x * 128 + wv * 32;      // output cols (of 512)

    const int m    = lane & 15;
    const int koff = (lane >> 4) * 2;                 // 0 or 2

    // A: tokens[m, k] = x[b, k, n0+m];  lane loads k = k0+koff, k0+koff+1
    const float* xa  = x + ((size_t)bz * C_ + koff) * N_ + (n0 + m);
    // B: W^T[k, o] = W[o, k];  lane owns columns o0+m and o0+16+m
    const float* wb0 = Wm + (size_t)(o0 + m)      * C_ + koff;
    const float* wb1 = Wm + (size_t)(o0 + 16 + m) * C_ + koff;

    v8f acc0 = {0.f,0.f,0.f,0.f,0.f,0.f,0.f,0.f};
    v8f acc1 = {0.f,0.f,0.f,0.f,0.f,0.f,0.f,0.f};
    #pragma unroll 4
    for (int k0 = 0; k0 < C_; k0 += 4) {
        v2f a;
        a.x = xa[(size_t)k0       * N_];
        a.y = xa[(size_t)(k0 + 1) * N_];
        v2f b0 = *(const v2f*)(wb0 + k0);
        v2f b1 = *(const v2f*)(wb1 + k0);
        acc0 = __builtin_amdgcn_wmma_f32_16x16x4_f32(
                   false, a, false, b0, (short)0, acc0, false, false);
        acc1 = __builtin_amdgcn_wmma_f32_16x16x4_f32(
                   false, a, false, b1, (short)0, acc1, false, false);
    }

    const float bv0  = bias[o0 + m];
    const float bv1  = bias[o0 + 16 + m];
    const int   rbase = n0 + ((lane >> 4) * 8);
    float* fp = featpos + ((size_t)bz * N_ + rbase) * C2_ + (o0 + m);
    #pragma unroll
    for (int v = 0; v < 8; ++v) {
        fp[(size_t)v * C2_]      = acc0[v] + bv0;
        fp[(size_t)v * C2_ + 16] = acc1[v] + bv1;
    }
}

// ---------------------------------------------------------------------------
// Kernel 2: in-place L2 row normalization of pos = featpos[..., 256:512]
// (one block per token)
// ---------------------------------------------------------------------------
__global__ __launch_bounds__(256) void norm_kernel(float* __restrict__ featpos)
{
    __shared__ float red[8];
    const int t = threadIdx.x;
    float* row  = featpos + (size_t)blockIdx.x * C2_ + C_;
    const float v = row[t];
    float ss = v * v;
    #pragma unroll
    for (int off = 16; off > 0; off >>= 1)
        ss += __shfl_xor(ss, off, 32);
    if ((t & 31) == 0) red[t >> 5] = ss;
    __syncthreads();
    float tot = 0.f;
    #pragma unroll
    for (int i = 0; i < 8; ++i) tot += red[i];
    const float s = 1.0f / fmaxf(sqrtf(tot), 1e-12f);
    row[t] = v * s;
}

// ---------------------------------------------------------------------------
// Kernel 3: fused  sim = pos @ pos^T  ->  top-16  ->  softmax  ->  gather.
// Block = 128 threads (4 waves) owns one (batch, 16-row tile).
// Column space (2304) processed in 9 chunks of 256 cols; each wave computes
// 4 column tiles as 2 pairs sharing the A fragment (1 LDS A load + 2 global
// B loads per 2 WMMAs).  Sim chunk staged in padded LDS; 16 threads maintain
// register-resident sorted top-16 lists.  Never materializes sim in HBM.
// ---------------------------------------------------------------------------
__global__ __launch_bounds__(128) void sim_topk_attn(
    const float* __restrict__ featpos,   // (B,N,2C): feat | normalized pos
    float* __restrict__ out)             // (B,C,N)
{
    __shared__ float posA[16 * 258];   // row-tile pos, padded vs bank conflicts
    __shared__ float simb[16 * 257];   // sim chunk, padded
    __shared__ float attn_s[16 * 16];
    __shared__ int   ids_s [16 * 16];

    const int tid  = threadIdx.x;
    const int lane = tid & 31;
    const int wv   = tid >> 5;
    const int bz   = blockIdx.y;
    const int row0 = blockIdx.x * 16;

    // stage 16 x 256 row-tile of normalized pos into LDS (coalesced)
    const float* prow = featpos + ((size_t)bz * N_ + row0) * C2_ + C_;
    #pragma unroll
    for (int i = 0; i < 32; ++i) {
        const int idx = i * 128 + tid;
        const int mm  = idx >> 8;
        const int cc  = idx & 255;
        posA[mm * 258 + cc] = prow[(size_t)mm * C2_ + cc];
    }
    __syncthreads();

    const int m    = lane & 15;
    const int koff = (lane >> 4) * 2;

    float bv[TOPK]; int bi[TOPK];
    #pragma unroll
    for (int k = 0; k < TOPK; ++k) { bv[k] = -3.0e38f; bi[k] = 0; }

    for (int cb = 0; cb < N_; cb += 256) {
        // ---- compute phase: each wave does 2 pairs of column tiles ----
        #pragma unroll
        for (int p = 0; p < 2; ++p) {
            const int tb0 = (wv * 4 + 2 * p) * 16;   // chunk-local col base
            const int n0  = cb + tb0;
            const float* pb0 = featpos + ((size_t)bz * N_ + (n0 + m))      * C2_ + C_ + koff;
            const float* pb1 = featpos + ((size_t)bz * N_ + (n0 + 16 + m)) * C2_ + C_ + koff;
            const float* pa  = &posA[m * 258 + koff];

            v8f acc0 = {0.f,0.f,0.f,0.f,0.f,0.f,0.f,0.f};
            v8f acc1 = {0.f,0.f,0.f,0.f,0.f,0.f,0.f,0.f};
            #pragma unroll 4
            for (int k0 = 0; k0 < C_; k0 += 4) {
                v2f a  = *(const v2f*)(pa  + k0);    // ds_load_b64, shared
                v2f b0 = *(const v2f*)(pb0 + k0);    // global 8B, L2-resident
                v2f b1 = *(const v2f*)(pb1 + k0);
                acc0 = __builtin_amdgcn_wmma_f32_16x16x4_f32(
                           false, a, false, b0, (short)0, acc0, false, false);
                acc1 = __builtin_amdgcn_wmma_f32_16x16x4_f32(
                           false, a, false, b1, (short)0, acc1, false, false);
            }
            const int mb = (lane >> 4) * 8;
            #pragma unroll
            for (int v = 0; v < 8; ++v) {
                simb[(mb + v) * 257 + tb0 + m]      = acc0[v];
                simb[(mb + v) * 257 + tb0 + 16 + m] = acc1[v];
            }
        }
        __syncthreads();

        // ---- top-k phase: thread t owns row t ----
        if (tid < 16) {
            for (int j = 0; j < 256; ++j) {
                const float v = simb[tid * 257 + j];
                if (v > bv[TOPK - 1]) {            // cheap reject
                    float cv = v; int ci = cb + j; // sorted insert (unrolled)
                    #pragma unroll
                    for (int k = 0; k < TOPK; ++k) {
                        if (cv > bv[k]) {
                            float tv = bv[k]; bv[k] = cv; cv = tv;
                            int   ti = bi[k]; bi[k] = ci; ci = ti;
                        }
                    }
                }
            }
        }
        __syncthreads();
    }

    // ---- softmax over top-16 (sorted desc, bv[0] is the max) ----
    if (tid < 16) {
        const float mx = bv[0];
        float e[TOPK]; float s = 0.f;
        #pragma unroll
        for (int k = 0; k < TOPK; ++k) { e[k] = __expf(bv[k] - mx); s += e[k]; }
        const float inv = 1.0f / s;
        #pragma unroll
        for (int k = 0; k < TOPK; ++k) {
            attn_s[tid * 16 + k] = e[k] * inv;
            ids_s [tid * 16 + k] = bi[k];
        }
    }
    __syncthreads();

    // ---- gather + weighted sum: out[b, c, row0+m] ----
    const float* fb = featpos + (size_t)bz * N_ * C2_;   // feat at offset 0
    float*       ob = out     + (size_t)bz * C_ * N_;
    for (int mm = 0; mm < 16; ++mm) {
        float a0 = 0.f, a1 = 0.f;
        #pragma unroll
        for (int k = 0; k < TOPK; ++k) {
            const float w   = attn_s[mm * 16 + k];
            const float* fr = fb + (size_t)ids_s[mm * 16 + k] * C2_;
            a0 += w * fr[tid];
            a1 += w * fr[tid + 128];
        }
        ob[(size_t)tid         * N_ + row0 + mm] = a0;
        ob[(size_t)(tid + 128) * N_ + row0 + mm] = a1;
    }
}

// ---------------------------------------------------------------------------
extern "C" void kernel_launch(void* const* d_in, const int* in_sizes, int n_in,
                              void* d_out, int out_size, void* d_ws, size_t ws_size,
                              hipStream_t stream)
{
    (void)in_sizes; (void)n_in; (void)out_size; (void)ws_size;

    const float* x    = (const float*)d_in[0];   // (B,C,H,W)
    const float* Wm   = (const float*)d_in[1];   // (2C,C)
    const float* bias = (const float*)d_in[2];   // (2C)
    float*       out  = (float*)d_out;           // (B,C,H,W)

    float* featpos = (float*)d_ws;               // B*N*2C f32 = 37.7 MB

    dim3 g1(C2_ / 128, N_ / 16, B_);             // (4,144,8)
    featpos_gemm<<<g1, 128, 0, stream>>>(x, Wm, bias, featpos);

    norm_kernel<<<dim3(B_ * N_), 256, 0, stream>>>(featpos);

    sim_topk_attn<<<dim3(N_ / 16, B_), 128, 0, stream>>>(featpos, out);
}